// HashSDF_27659589386593
// MI455X (gfx1250) — compile-verified
//
#include <hip/hip_runtime.h>
#include <stdint.h>

// ---- Instant-NGP constants (match reference) ----
#define NLEV 12
#define TBLSZ (1u << 19)
#define TMASK (TBLSZ - 1u)
#define PRIME1 2654435761u
#define PRIME2 805459861u
#define KDIM 48            // 12 levels * 4 feats
#define ODIM 28            // output columns
#define BLK 256            // threads per block (8 wave32)
#define ROWSTRIDE 52       // LDS row stride in floats (bank-conflict-free)

typedef __attribute__((ext_vector_type(2))) float v2f;
typedef __attribute__((ext_vector_type(8))) float v8f;

// floor(16 * (128^(1/11))^l), l = 0..11  (float64 host math reproduced)
__device__ __constant__ float c_res[NLEV] = {
    16.f, 24.f, 38.f, 60.f, 93.f, 145.f, 225.f, 350.f, 545.f, 847.f, 1317.f, 2048.f
};

__global__ __launch_bounds__(BLK)
void hashgrid_wmma_kernel(const float* __restrict__ xyz,     // [N,3]
                          const float* __restrict__ tables,  // [12, 2^19, 4]
                          const float* __restrict__ W,       // [48, 28] row-major
                          float* __restrict__ out,           // [N, 28]
                          int Npts)
{
    __shared__ float sfeat[BLK * ROWSTRIDE];   // 53,248 B

    const int tid  = threadIdx.x;
    const int lane = tid & 31;
    const int col  = lane & 15;                 // N-index within a 16-wide tile
    const int krow = (lane < 16) ? 0 : 2;       // K sub-offset per f32 WMMA A/B layout
    const int p    = blockIdx.x * BLK + tid;

    // ---------------- Preload B fragments (W) into registers ----------------
    // B is 4x16 per wmma step: lanes 0-15 hold K=4k,4k+1 ; lanes 16-31 hold K=4k+2,4k+3.
    v2f bfrag[2][NLEV];
    #pragma unroll
    for (int n = 0; n < 2; ++n) {
        const int c = n * 16 + col;
        const bool valid = (c < ODIM);
        #pragma unroll
        for (int k = 0; k < NLEV; ++k) {
            const int r = 4 * k + krow;
            bfrag[n][k].x = valid ? W[r * ODIM + c]       : 0.0f;
            bfrag[n][k].y = valid ? W[(r + 1) * ODIM + c] : 0.0f;
        }
    }

    // ---------------- Phase 1: hash-grid embedding -> LDS ----------------
    float px = 0.f, py = 0.f, pz = 0.f;
    if (p < Npts) {
        px = xyz[p * 3 + 0];
        py = xyz[p * 3 + 1];
        pz = xyz[p * 3 + 2];
    }

    #pragma unroll
    for (int l = 0; l < NLEV; ++l) {
        const float res = c_res[l];
        const float fx = px * res, fy = py * res, fz = pz * res;
        const float flx = floorf(fx), fly = floorf(fy), flz = floorf(fz);
        const float tx = fx - flx, ty = fy - fly, tz = fz - flz;
        const uint32_t ix = (uint32_t)flx, iy = (uint32_t)fly, iz = (uint32_t)flz;

        const float    wx[2] = {1.0f - tx, tx};
        const float    wy[2] = {1.0f - ty, ty};
        const float    wz[2] = {1.0f - tz, tz};
        const uint32_t hx[2] = {ix, ix + 1u};                               // prime P0 = 1
        const uint32_t hy[2] = {iy * PRIME1, (iy + 1u) * PRIME1};
        const uint32_t hz[2] = {iz * PRIME2, (iz + 1u) * PRIME2};

        const float4* __restrict__ tbl =
            (const float4*)tables + (size_t)l * TBLSZ;

        float a0 = 0.f, a1 = 0.f, a2 = 0.f, a3 = 0.f;
        #pragma unroll
        for (int ci = 0; ci < 2; ++ci) {
            #pragma unroll
            for (int cj = 0; cj < 2; ++cj) {
                #pragma unroll
                for (int ck = 0; ck < 2; ++ck) {
                    const uint32_t idx = (hx[ci] ^ hy[cj] ^ hz[ck]) & TMASK;
                    const float4 f = tbl[idx];                 // global_load_b128 gather (L2-resident)
                    const float w = wx[ci] * wy[cj] * wz[ck];
                    a0 = fmaf(w, f.x, a0);
                    a1 = fmaf(w, f.y, a1);
                    a2 = fmaf(w, f.z, a2);
                    a3 = fmaf(w, f.w, a3);
                }
            }
        }
        float* s = &sfeat[tid * ROWSTRIDE + l * 4];
        s[0] = a0; s[1] = a1; s[2] = a2; s[3] = a3;
    }

    __syncthreads();

    // ---------------- Phase 2: [16,48] x [48,28] via V_WMMA_F32_16X16X4_F32 ----------------
    const int wv = tid >> 5;          // wave id 0..7; each wave owns 32 points = 2 M-tiles
    #pragma unroll
    for (int mt = 0; mt < 2; ++mt) {
        const int rbase = wv * 32 + mt * 16;     // local point base of this 16-row tile
        const int arow  = rbase + col;           // lanes 0-15 and 16-31 map to same M rows

        // A fragments: lane holds feats[arow][4k+krow .. 4k+krow+1]
        v2f afrag[NLEV];
        #pragma unroll
        for (int k = 0; k < NLEV; ++k) {
            const float* sp = &sfeat[arow * ROWSTRIDE + 4 * k + krow];
            afrag[k] = *(const v2f*)sp;          // ds_load_2addr_b64, conflict-free
        }

        const int gp0      = blockIdx.x * BLK + rbase;
        const bool fullTile = (gp0 + 16) <= Npts;          // uniform per tile
        const int r0       = gp0 + ((lane < 16) ? 0 : 8);  // first of this lane-half's 8 rows

        #pragma unroll
        for (int n = 0; n < 2; ++n) {
            v8f acc = {};
            #pragma unroll
            for (int k = 0; k < NLEV; ++k) {
                acc = __builtin_amdgcn_wmma_f32_16x16x4_f32(
                    /*neg_a=*/false, afrag[k],
                    /*neg_b=*/false, bfrag[n][k],
                    /*c_mod=*/(short)0, acc,
                    /*reuse_a=*/false, /*reuse_b=*/false);
            }
            // D layout: VGPR j -> (M=j, N=lane) lanes 0-15 ; (M=j+8, N=lane-16) lanes 16-31
            const int c = n * 16 + col;
            if (c < ODIM) {
                float* __restrict__ po = out + (size_t)r0 * ODIM + c;
                if (fullTile) {
                    // hot path: 8 stores, immediate offsets, no per-row predication
                    #pragma unroll
                    for (int j = 0; j < 8; ++j) po[j * ODIM] = acc[j];
                } else {
                    #pragma unroll
                    for (int j = 0; j < 8; ++j)
                        if (r0 + j < Npts) po[j * ODIM] = acc[j];
                }
            }
        }
    }
}

extern "C" void kernel_launch(void* const* d_in, const int* in_sizes, int n_in,
                              void* d_out, int out_size, void* d_ws, size_t ws_size,
                              hipStream_t stream) {
    const float* xyz    = (const float*)d_in[0];   // [N,3]
    const float* tables = (const float*)d_in[1];   // [12, 2^19, 4]
    const float* W      = (const float*)d_in[2];   // [48, 28]
    float* out          = (float*)d_out;           // [N, 28]

    const int N = in_sizes[0] / 3;
    const int grid = (N + BLK - 1) / BLK;
    hashgrid_wmma_kernel<<<grid, BLK, 0, stream>>>(xyz, tables, W, out, N);
}